// HypAggOld_47107201303151
// MI455X (gfx1250) — compile-verified
//
#include <hip/hip_runtime.h>
#include <stdint.h>

#define D_FEAT 128
#define CHUNK 1024                         // edges staged per buffer
#define CHUNK_BYTES (CHUNK * 4)            // bytes per staged array (4096)
#define BLOCK_THREADS 256                  // 8 wave32 per block
#define WAVES_PER_BLOCK (BLOCK_THREADS / 32)
#define EDGES_PER_WAVE (CHUNK / WAVES_PER_BLOCK)   // 128

static __device__ __forceinline__ int imin(int a, int b) { return a < b ? a : b; }

static __device__ __forceinline__ float wave_reduce_sum(float v) {
#pragma unroll
  for (int m = 16; m > 0; m >>= 1) v += __shfl_xor(v, m, 32);
  return v;
}

// ---------------- Kernel 1: logmap0 + zero the accumulator (d_out) ------------
// One wave per node row; lane l owns features [4l, 4l+4).
__global__ void k_logmap0_zero(const float* __restrict__ x,
                               float* __restrict__ xt,
                               float* __restrict__ out, int n) {
  int node = (int)((blockIdx.x * blockDim.x + threadIdx.x) >> 5);
  int lane = threadIdx.x & 31;
  if (node >= n) return;
  size_t rowoff = (size_t)node * D_FEAT;
  float4 v = ((const float4*)(x + rowoff))[lane];
  float ss = v.x * v.x + v.y * v.y + v.z * v.z + v.w * v.w;
  ss = wave_reduce_sum(ss);
  float pn = fmaxf(sqrtf(ss), 1e-15f);          // ||x||, clamped
  float y  = fminf(pn, 1.0f - 1e-7f);           // artanh clip (sqrt_c = 1)
  float s  = (0.5f * logf((1.0f + y) / (1.0f - y))) / pn;  // artanh(y)/pn
  float4 o; o.x = s * v.x; o.y = s * v.y; o.z = s * v.z; o.w = s * v.w;
  ((float4*)(xt + rowoff))[lane] = o;
  float4 z; z.x = 0.f; z.y = 0.f; z.z = 0.f; z.w = 0.f;
  ((float4*)(out + rowoff))[lane] = z;          // accumulator init
}

// ---------------- Kernel 2: staged COO scatter-add ----------------------------
// Edge metadata (row / col / w) is DMA'd into LDS with the CDNA5 async path
// (GLOBAL_LOAD_ASYNC_TO_LDS_B128, ASYNCcnt), double-buffered across chunks.
// Each wave then walks 128 edges: one float4 gather from x_tangent per lane
// plus 4 fire-and-forget global_atomic_add_f32.

static __device__ __forceinline__ void stage_chunk(uint32_t lds_base,
                                                   const uint32_t* __restrict__ rows,
                                                   const uint32_t* __restrict__ cols,
                                                   const uint32_t* __restrict__ wts,
                                                   int c0, int E, int t) {
  int elem = c0 + t * 4;                 // 4 u32 (16B) per thread per array
  int cl = elem;                         // clamp tail so we never read OOB
  if (cl > E - 4) cl = E - 4;
  if (cl < 0) cl = 0;
  uint32_t goff = (uint32_t)cl * 4u;     // byte offset into each edge array
  uint32_t l = lds_base + (uint32_t)t * 16u;
  asm volatile("global_load_async_to_lds_b128 %0, %1, %2"
               :: "v"(l), "v"(goff), "s"(rows) : "memory");
  asm volatile("global_load_async_to_lds_b128 %0, %1, %2"
               :: "v"(l + (uint32_t)CHUNK_BYTES), "v"(goff), "s"(cols) : "memory");
  asm volatile("global_load_async_to_lds_b128 %0, %1, %2"
               :: "v"(l + (uint32_t)(2 * CHUNK_BYTES)), "v"(goff), "s"(wts) : "memory");
}

__global__ void k_spmm_scatter(const float* __restrict__ xt,
                               const uint32_t* __restrict__ rows,
                               const uint32_t* __restrict__ cols,
                               const uint32_t* __restrict__ wts,
                               float* __restrict__ out, int E) {
  __shared__ uint32_t sb[2][3 * CHUNK];  // 24 KB: {row|col|w} x double buffer
  const int t = threadIdx.x;
  const int lane = t & 31;
  const int wave = t >> 5;
  uint32_t lb0 = (uint32_t)(uintptr_t)(&sb[0][0]);
  uint32_t lb1 = (uint32_t)(uintptr_t)(&sb[1][0]);

  int numChunks = (E + CHUNK - 1) / CHUNK;
  int c = (int)blockIdx.x;
  if (c >= numChunks) return;

  stage_chunk(lb0, rows, cols, wts, c * CHUNK, E, t);
  int buf = 0;
  for (;;) {
    asm volatile("s_wait_asynccnt 0x0" ::: "memory");   // my async loads landed
    __syncthreads();                                    // everyone's landed

    int cn = c + (int)gridDim.x;
    if (cn < numChunks)                                 // prefetch next chunk
      stage_chunk(buf ? lb0 : lb1, rows, cols, wts, cn * CHUNK, E, t);

    const uint32_t* sR = &sb[buf][0];
    const uint32_t* sC = &sb[buf][CHUNK];
    const uint32_t* sW = &sb[buf][2 * CHUNK];
    int cnt = imin(CHUNK, E - c * CHUNK);
    int beg = wave * EDGES_PER_WAVE;
    int end = imin(beg + EDGES_PER_WAVE, cnt);
#pragma unroll 4
    for (int e = beg; e < end; ++e) {
      uint32_t r  = sR[e];                              // LDS broadcast reads
      uint32_t cc = sC[e];
      float    wt = __uint_as_float(sW[e]);
      float4 v = ((const float4*)(xt + (size_t)cc * D_FEAT))[lane];
      float* dst = out + (size_t)r * D_FEAT + lane * 4;
      atomicAdd(dst + 0, wt * v.x);                     // no-return f32 atomics
      atomicAdd(dst + 1, wt * v.y);
      atomicAdd(dst + 2, wt * v.z);
      atomicAdd(dst + 3, wt * v.w);
    }
    __syncthreads();                                    // safe to overwrite buf
    if (cn >= numChunks) break;
    c = cn;
    buf ^= 1;
  }
}

// ---------------- Kernel 3: expmap0 + projection, in place on d_out -----------
__global__ void k_expmap0_proj(float* __restrict__ out, int n) {
  int node = (int)((blockIdx.x * blockDim.x + threadIdx.x) >> 5);
  int lane = threadIdx.x & 31;
  if (node >= n) return;
  size_t rowoff = (size_t)node * D_FEAT;
  float4 v = ((float4*)(out + rowoff))[lane];
  float ss = wave_reduce_sum(v.x * v.x + v.y * v.y + v.z * v.z + v.w * v.w);
  float un = fmaxf(sqrtf(ss), 1e-15f);
  float th = tanhf(un);
  float s1 = th / un;                                   // expmap0 scale (sqrt_c=1)
  float4 y; y.x = s1 * v.x; y.y = s1 * v.y; y.z = s1 * v.z; y.w = s1 * v.w;
  float sy = wave_reduce_sum(y.x * y.x + y.y * y.y + y.z * y.z + y.w * y.w);
  float yn = fmaxf(sqrtf(sy), 1e-15f);
  const float maxnorm = 1.0f - 4e-3f;                   // (1 - PROJ_EPS)/sqrt_c
  float s2 = (yn > maxnorm) ? (maxnorm / yn) : 1.0f;
  float4 o; o.x = s2 * y.x; o.y = s2 * y.y; o.z = s2 * y.z; o.w = s2 * y.w;
  ((float4*)(out + rowoff))[lane] = o;
}

// ---------------- Launcher ----------------------------------------------------
extern "C" void kernel_launch(void* const* d_in, const int* in_sizes, int n_in,
                              void* d_out, int out_size, void* d_ws, size_t ws_size,
                              hipStream_t stream) {
  const float*    x  = (const float*)d_in[0];
  const uint32_t* ew = (const uint32_t*)d_in[1];   // float bits, reinterpreted in LDS
  const uint32_t* er = (const uint32_t*)d_in[2];
  const uint32_t* ec = (const uint32_t*)d_in[3];
  float* out = (float*)d_out;
  float* xt  = (float*)d_ws;                       // N*D floats of scratch

  int N = in_sizes[0] / D_FEAT;
  int E = in_sizes[1];

  int blocksRow = (N + WAVES_PER_BLOCK - 1) / WAVES_PER_BLOCK;
  k_logmap0_zero<<<blocksRow, BLOCK_THREADS, 0, stream>>>(x, xt, out, N);

  int numChunks = (E + CHUNK - 1) / CHUNK;
  int blocksB = numChunks < 16384 ? numChunks : 16384;
  k_spmm_scatter<<<blocksB, BLOCK_THREADS, 0, stream>>>(xt, er, ec, ew, out, E);

  k_expmap0_proj<<<blocksRow, BLOCK_THREADS, 0, stream>>>(out, N);
}